// VQVAE_32384053411993
// MI455X (gfx1250) — compile-verified
//
#include <hip/hip_runtime.h>

typedef float f32x2 __attribute__((ext_vector_type(2)));
typedef float f32x8 __attribute__((ext_vector_type(8)));

#define N_TOT   16384   // B*H*W
#define D_CODE  1024
#define KCODES  2048
#define C_IN    1024
#define HW      1024

// --------------------------------------------------------------------------
// CDNA5 async global->LDS copy (ASYNCcnt path). LDS operand is the low 32
// bits of the generic pointer (flat->LDS addresses truncate per ISA 10.2).
// --------------------------------------------------------------------------
#define USE_ASYNC_LDS 1

__device__ __forceinline__ void ldsg16(const float* __restrict__ g, float* l) {
#if USE_ASYNC_LDS
  asm volatile("global_load_async_to_lds_b128 %0, %1, off"
               :
               : "v"((unsigned int)(unsigned long long)l),
                 "v"((unsigned long long)g)
               : "memory");
#else
  *(float4*)l = *(const float4*)g;
#endif
}

__device__ __forceinline__ void lds_async_wait() {
#if USE_ASYNC_LDS
  asm volatile("s_wait_asynccnt 0x0" ::: "memory");
#endif
}

// ---------------------------------------------------------------------------
// GEMM1: xf_t[d][n] = sum_c x[b,c,hw] * conv_w[d,c] + bias[d]
// 64(n) x 128(d) macro-tile, 8 waves of 32x32, double-buffered LDS tiles.
// A tile filled with async global->LDS (row copy); B needs a transpose so it
// goes global->VGPR->ds_store, overlapped with the WMMAs.
// ---------------------------------------------------------------------------
__global__ __launch_bounds__(256) void vq_proj_gemm(
    const float* __restrict__ x, const float* __restrict__ w,
    const float* __restrict__ bias, float* __restrict__ xf_t)
{
  __shared__ __align__(16) float As[2][16 * 64];    // [k][m]
  __shared__ __align__(16) float Bs[2][16 * 140];   // [k][n], padded stride
  __shared__ __align__(16) float Cs[64 * 132];      // staged C

  const int t    = threadIdx.x;
  const int lane = t & 31;
  const int wv   = t >> 5;
  const int mi   = wv & 1;       // 2 row halves of 32
  const int ni   = wv >> 1;      // 4 col quarters of 32
  const int l15  = lane & 15;
  const int lhi  = lane >> 4;

  const int nbase = blockIdx.y * 64;
  const int dbase = blockIdx.x * 128;
  const int bimg  = nbase >> 10;
  const int hw0   = nbase & 1023;

  const int akk = t >> 4;            // A fill: k row (0..15)
  const int am  = (t & 15) * 4;      // A fill: m col *4
  const int bdn = t >> 2;            // B fill: d col (0..63; +64 second half)
  const int bck = (t & 3) * 4;       // B fill: c row *4

  const float* xA = x + ((size_t)bimg * C_IN) * HW + hw0;
  const float* wB = w + (size_t)dbase * C_IN;

  // ---- prologue: fill buffer 0 for kb = 0 ----
  ldsg16(xA + (size_t)akk * HW + am, &As[0][akk * 64 + am]);
  {
    float4 w0 = *(const float4*)(wB + (size_t)bdn * C_IN + bck);
    float4 w1 = *(const float4*)(wB + (size_t)(bdn + 64) * C_IN + bck);
    Bs[0][(bck + 0) * 140 + bdn] = w0.x;  Bs[0][(bck + 1) * 140 + bdn] = w0.y;
    Bs[0][(bck + 2) * 140 + bdn] = w0.z;  Bs[0][(bck + 3) * 140 + bdn] = w0.w;
    Bs[0][(bck + 0) * 140 + bdn + 64] = w1.x;  Bs[0][(bck + 1) * 140 + bdn + 64] = w1.y;
    Bs[0][(bck + 2) * 140 + bdn + 64] = w1.z;  Bs[0][(bck + 3) * 140 + bdn + 64] = w1.w;
  }
  lds_async_wait();
  __syncthreads();

  f32x8 acc00 = {}, acc01 = {}, acc10 = {}, acc11 = {};

  const int ar0 = mi * 32 + l15, ar1 = mi * 32 + 16 + l15;
  const int bc0 = ni * 32 + l15, bc1 = ni * 32 + 16 + l15;

  for (int kb = 0; kb < C_IN; kb += 16) {
    const int buf = (kb >> 4) & 1;
    const int nxt = buf ^ 1;
    const bool more = (kb + 16) < C_IN;
    float4 nw0, nw1;
    if (more) {
      ldsg16(xA + (size_t)(kb + 16 + akk) * HW + am, &As[nxt][akk * 64 + am]);
      nw0 = *(const float4*)(wB + (size_t)bdn * C_IN + kb + 16 + bck);
      nw1 = *(const float4*)(wB + (size_t)(bdn + 64) * C_IN + kb + 16 + bck);
    }
#pragma unroll
    for (int ks = 0; ks < 4; ++ks) {
      const int kp = ks * 4 + lhi * 2;
      f32x2 a0, a1, b0, b1;
      a0.x = As[buf][kp * 64 + ar0];       a0.y = As[buf][(kp + 1) * 64 + ar0];
      a1.x = As[buf][kp * 64 + ar1];       a1.y = As[buf][(kp + 1) * 64 + ar1];
      b0.x = Bs[buf][kp * 140 + bc0];      b0.y = Bs[buf][(kp + 1) * 140 + bc0];
      b1.x = Bs[buf][kp * 140 + bc1];      b1.y = Bs[buf][(kp + 1) * 140 + bc1];
      acc00 = __builtin_amdgcn_wmma_f32_16x16x4_f32(false, a0, false, b0, (short)0, acc00, false, false);
      acc01 = __builtin_amdgcn_wmma_f32_16x16x4_f32(false, a0, false, b1, (short)0, acc01, false, false);
      acc10 = __builtin_amdgcn_wmma_f32_16x16x4_f32(false, a1, false, b0, (short)0, acc10, false, false);
      acc11 = __builtin_amdgcn_wmma_f32_16x16x4_f32(false, a1, false, b1, (short)0, acc11, false, false);
    }
    if (more) {
      Bs[nxt][(bck + 0) * 140 + bdn] = nw0.x;  Bs[nxt][(bck + 1) * 140 + bdn] = nw0.y;
      Bs[nxt][(bck + 2) * 140 + bdn] = nw0.z;  Bs[nxt][(bck + 3) * 140 + bdn] = nw0.w;
      Bs[nxt][(bck + 0) * 140 + bdn + 64] = nw1.x;  Bs[nxt][(bck + 1) * 140 + bdn + 64] = nw1.y;
      Bs[nxt][(bck + 2) * 140 + bdn + 64] = nw1.z;  Bs[nxt][(bck + 3) * 140 + bdn + 64] = nw1.w;
    }
    lds_async_wait();
    __syncthreads();
  }

  // ---- stage C through LDS for coalesced transposed store ----
#pragma unroll
  for (int r = 0; r < 8; ++r) {
    const int r0 = mi * 32 + lhi * 8 + r;
    const int r1 = r0 + 16;
    Cs[r0 * 132 + bc0] = acc00[r];
    Cs[r0 * 132 + bc1] = acc01[r];
    Cs[r1 * 132 + bc0] = acc10[r];
    Cs[r1 * 132 + bc1] = acc11[r];
  }
  __syncthreads();
  for (int i = t; i < 64 * 128; i += 256) {
    const int rn = i & 63;
    const int cd = i >> 6;
    xf_t[(size_t)(dbase + cd) * N_TOT + nbase + rn] = Cs[rn * 132 + cd] + bias[dbase + cd];
  }
}

// ---------------------------------------------------------------------------
// cb2[k] = sum_d codebook[d][k]^2
// ---------------------------------------------------------------------------
__global__ __launch_bounds__(256) void vq_cb2(const float* __restrict__ cb,
                                              float* __restrict__ cb2)
{
  const int k = blockIdx.x * 256 + threadIdx.x;
  float s = 0.0f;
  for (int d = 0; d < D_CODE; ++d) {
    const float v = cb[(size_t)d * KCODES + k];
    s += v * v;
  }
  cb2[k] = s;
}

// ---------------------------------------------------------------------------
// GEMM2 + fused argmin: score[n][k] = cb2[k] - 2 * sum_d xf_t[d][n]*cb[d][k]
// 64-row A strip resident in LDS (256 KB of the WGP's 320 KB), then stream
// 16 code tiles of 128 with double-buffered async B fills.
// ---------------------------------------------------------------------------
__global__ __launch_bounds__(256) void vq_dist_argmin(
    const float* __restrict__ xf_t, const float* __restrict__ cb,
    const float* __restrict__ cb2, int* __restrict__ idx)
{
  __shared__ __align__(16) float Ax[D_CODE * 64];   // [kd][n] strip: 256 KB
  __shared__ __align__(16) float Bs[2][16 * 140];
  __shared__ float rv[64 * 4];
  __shared__ int   ri[64 * 4];

  const int t    = threadIdx.x;
  const int lane = t & 31;
  const int wv   = t >> 5;
  const int mi   = wv & 1;
  const int ni   = wv >> 1;
  const int l15  = lane & 15;
  const int lhi  = lane >> 4;
  const int nbase = blockIdx.x * 64;

  // ---- preload whole A strip (once) via async copies ----
  for (int i = 0; i < 64; ++i) {
    const int e  = i * 256 + t;        // 0..16383 float4 tiles
    const int kd = e >> 4;
    const int n4 = (e & 15) * 4;
    ldsg16(xf_t + (size_t)kd * N_TOT + nbase + n4, &Ax[kd * 64 + n4]);
  }
  // ---- prologue B fill: ct = 0, kd = 0 ----
#pragma unroll
  for (int j = 0; j < 2; ++j) {
    const int e   = t + j * 256;
    const int bkk = e >> 5;
    const int bn4 = (e & 31) * 4;
    ldsg16(cb + (size_t)bkk * KCODES + bn4, &Bs[0][bkk * 140 + bn4]);
  }
  lds_async_wait();
  __syncthreads();

  float bv[16]; int bix[16];
#pragma unroll
  for (int r = 0; r < 16; ++r) { bv[r] = __builtin_inff(); bix[r] = 0; }

  const int ar0 = mi * 32 + l15, ar1 = mi * 32 + 16 + l15;
  const int bc0 = ni * 32 + l15, bc1 = ni * 32 + 16 + l15;

  for (int ct = 0; ct < KCODES; ct += 128) {
    f32x8 acc00 = {}, acc01 = {}, acc10 = {}, acc11 = {};
    for (int kd = 0; kd < D_CODE; kd += 16) {
      const int buf = (kd >> 4) & 1;
      const int nxt = buf ^ 1;
      const bool morek  = (kd + 16) < D_CODE;
      const bool morect = (ct + 128) < KCODES;
      if (morek || morect) {
        const int nkd = morek ? kd + 16 : 0;
        const int nct = morek ? ct : ct + 128;
#pragma unroll
        for (int j = 0; j < 2; ++j) {
          const int e   = t + j * 256;
          const int bkk = e >> 5;
          const int bn4 = (e & 31) * 4;
          ldsg16(cb + (size_t)(nkd + bkk) * KCODES + nct + bn4,
                 &Bs[nxt][bkk * 140 + bn4]);
        }
      }
#pragma unroll
      for (int ks = 0; ks < 4; ++ks) {
        const int kp = ks * 4 + lhi * 2;
        f32x2 a0, a1, b0, b1;
        a0.x = Ax[(kd + kp) * 64 + ar0];     a0.y = Ax[(kd + kp + 1) * 64 + ar0];
        a1.x = Ax[(kd + kp) * 64 + ar1];     a1.y = Ax[(kd + kp + 1) * 64 + ar1];
        b0.x = Bs[buf][kp * 140 + bc0];      b0.y = Bs[buf][(kp + 1) * 140 + bc0];
        b1.x = Bs[buf][kp * 140 + bc1];      b1.y = Bs[buf][(kp + 1) * 140 + bc1];
        acc00 = __builtin_amdgcn_wmma_f32_16x16x4_f32(false, a0, false, b0, (short)0, acc00, false, false);
        acc01 = __builtin_amdgcn_wmma_f32_16x16x4_f32(false, a0, false, b1, (short)0, acc01, false, false);
        acc10 = __builtin_amdgcn_wmma_f32_16x16x4_f32(false, a1, false, b0, (short)0, acc10, false, false);
        acc11 = __builtin_amdgcn_wmma_f32_16x16x4_f32(false, a1, false, b1, (short)0, acc11, false, false);
      }
      lds_async_wait();
      __syncthreads();
    }
    // fused argmin update (||x||^2 constant per row: irrelevant for argmin)
    const int cc0 = ct + ni * 32 + l15;
    const int cc1 = cc0 + 16;
    const float q0 = cb2[cc0];
    const float q1 = cb2[cc1];
#pragma unroll
    for (int r = 0; r < 8; ++r) {
      float s;
      s = q0 - 2.0f * acc00[r]; if (s < bv[r])     { bv[r] = s;     bix[r] = cc0; }
      s = q1 - 2.0f * acc01[r]; if (s < bv[r])     { bv[r] = s;     bix[r] = cc1; }
      s = q0 - 2.0f * acc10[r]; if (s < bv[8 + r]) { bv[8 + r] = s; bix[8 + r] = cc0; }
      s = q1 - 2.0f * acc11[r]; if (s < bv[8 + r]) { bv[8 + r] = s; bix[8 + r] = cc1; }
    }
  }

  // reduce across the 16 lanes sharing a row-set (different code columns)
#pragma unroll
  for (int off = 8; off >= 1; off >>= 1) {
#pragma unroll
    for (int r = 0; r < 16; ++r) {
      const float ov = __shfl_xor(bv[r], off, 32);
      const int   oi = __shfl_xor(bix[r], off, 32);
      if (ov < bv[r] || (ov == bv[r] && oi < bix[r])) { bv[r] = ov; bix[r] = oi; }
    }
  }
  if (l15 == 0) {
#pragma unroll
    for (int ai = 0; ai < 2; ++ai)
#pragma unroll
      for (int r = 0; r < 8; ++r) {
        const int row = mi * 32 + ai * 16 + lhi * 8 + r;
        rv[row * 4 + ni] = bv[ai * 8 + r];
        ri[row * 4 + ni] = bix[ai * 8 + r];
      }
  }
  __syncthreads();
  if (t < 64) {
    float best = rv[t * 4];
    int   bi   = ri[t * 4];
#pragma unroll
    for (int g = 1; g < 4; ++g) {
      const float v = rv[t * 4 + g];
      const int   i = ri[t * 4 + g];
      if (v < best || (v == best && i < bi)) { best = v; bi = i; }
    }
    idx[nbase + t] = bi;
  }
}

// ---------------------------------------------------------------------------
// Gather quant, write quant_st = xp + (q - xp), per-block partial of (q-xp)^2
// ---------------------------------------------------------------------------
__global__ __launch_bounds__(256) void vq_gather_diff(
    const float* __restrict__ cb, const float* __restrict__ xf_t,
    const int* __restrict__ idx, float* __restrict__ out,
    float* __restrict__ partial)
{
  const int d = blockIdx.x;   // 0..1023
  const int b = blockIdx.y;   // 0..15
  const float* cbrow = cb + (size_t)d * KCODES;
  float ssum = 0.0f;
#pragma unroll
  for (int i = 0; i < 4; ++i) {
    const int hw = threadIdx.x + i * 256;
    const int n  = b * 1024 + hw;
    const float q  = cbrow[idx[n]];
    const float xp = xf_t[(size_t)d * N_TOT + n];
    const float df = q - xp;
    out[((size_t)b * D_CODE + d) * HW + hw] = xp + df;  // straight-through
    ssum += df * df;
  }
  __shared__ float red[256];
  red[threadIdx.x] = ssum;
  __syncthreads();
  for (int s = 128; s > 0; s >>= 1) {
    if (threadIdx.x < s) red[threadIdx.x] += red[threadIdx.x + s];
    __syncthreads();
  }
  if (threadIdx.x == 0) partial[b * D_CODE + d] = red[0];
}

// ---------------------------------------------------------------------------
// Finalize: deterministic reduce of 16384 partials -> diff; emit indices.
// ---------------------------------------------------------------------------
__global__ __launch_bounds__(256) void vq_finalize(
    const float* __restrict__ partial, const int* __restrict__ idx,
    float* __restrict__ out)
{
  __shared__ float red[256];
  float s = 0.0f;
  for (int i = threadIdx.x; i < 16384; i += 256) s += partial[i];
  red[threadIdx.x] = s;
  __syncthreads();
  for (int st = 128; st > 0; st >>= 1) {
    if (threadIdx.x < st) red[threadIdx.x] += red[threadIdx.x + st];
    __syncthreads();
  }
  if (threadIdx.x == 0)
    out[(size_t)16777216] = red[0] * (1.0f / 16777216.0f);
  for (int i = threadIdx.x; i < 16384; i += 256)
    out[(size_t)16777217 + i] = (float)idx[i];
}

extern "C" void kernel_launch(void* const* d_in, const int* in_sizes, int n_in,
                              void* d_out, int out_size, void* d_ws, size_t ws_size,
                              hipStream_t stream) {
  const float* x    = (const float*)d_in[0];  // (16,1024,32,32)
  const float* w    = (const float*)d_in[1];  // (1024,1024) [d][c]
  const float* bias = (const float*)d_in[2];  // (1024,)
  const float* cb   = (const float*)d_in[3];  // (1024,2048) [d][k]
  float* out = (float*)d_out;

  char* ws = (char*)d_ws;
  float* xf_t    = (float*)ws;                                     // 64 MB: [d][n]
  float* cb2     = (float*)(ws + (size_t)67108864);                // 8 KB
  int*   idx     = (int*)  (ws + (size_t)67108864 + 8192);         // 64 KB
  float* partial = (float*)(ws + (size_t)67108864 + 8192 + 65536); // 64 KB

  vq_proj_gemm<<<dim3(8, 256), 256, 0, stream>>>(x, w, bias, xf_t);
  vq_cb2<<<8, 256, 0, stream>>>(cb, cb2);
  vq_dist_argmin<<<256, 256, 0, stream>>>(xf_t, cb, cb2, idx);
  vq_gather_diff<<<dim3(1024, 16), 256, 0, stream>>>(cb, xf_t, idx, out, partial);
  vq_finalize<<<1, 256, 0, stream>>>(partial, idx, out);
}